// Attention_82841329205561
// MI455X (gfx1250) — compile-verified
//
#include <hip/hip_runtime.h>
#include <hip/hip_bf16.h>
#include <cstdint>

// ---------------- problem constants ----------------
static constexpr int B  = 32;
static constexpr int C  = 128;
static constexpr int CQ = 32;     // C/4
static constexpr int L  = 8192;
static constexpr float EPS = 1e-5f;

static constexpr int TILE   = 64;         // L-positions per block in GEMM kernels
static constexpr int STRIDE = 72;         // LDS row stride (2*72 mod 64 banks = 16 -> halves hit disjoint banks)
static constexpr int SUB    = TILE / 16;  // 16-wide N subtiles per wave

typedef float v2f __attribute__((ext_vector_type(2)));
typedef float v8f __attribute__((ext_vector_type(8)));

#if __has_builtin(__builtin_amdgcn_tensor_load_to_lds) && __has_builtin(__builtin_amdgcn_s_wait_tensorcnt)
#define USE_TDM 1
typedef unsigned int v4u __attribute__((ext_vector_type(4)));
typedef int          v8i __attribute__((ext_vector_type(8)));
typedef int          v4i __attribute__((ext_vector_type(4)));

// Issue a TDM 2D tile load: tile_dim0=64 cols (dwords), tile_dim1=128 rows,
// row stride = L elements, fp32, LDS padding: +8 dwords after every 64 dwords
// stored  ->  rows land at LDS stride 72 floats (matches xs[] layout).
static __device__ __forceinline__ void tdm_load_tile(const float* gsrc, uint32_t lds_byte_off) {
  const uint64_t ga = (uint64_t)(uintptr_t)gsrc;
  v4u g0;
  g0.x = 1u;                                              // count=1 (valid descriptor)
  g0.y = lds_byte_off;                                    // lds_addr (bytes)
  g0.z = (uint32_t)ga;                                    // global_addr[31:0]
  g0.w = (uint32_t)((ga >> 32) & 0x1FFFFFFu) | (2u << 30);// global_addr[56:32] | type=2
  v8i g1;
  g1[0] = (int)((2u << 16)      // data_size = 4 bytes
              | (1u << 20)      // pad_enable
              | (5u << 22)      // pad_interval: 64 dwords
              | (7u << 25));    // pad_amount : 8 dwords
  g1[1] = (int)((uint32_t)L << 16);   // tensor_dim0[15:0] in [31:16]
  g1[2] = (int)((uint32_t)C << 16);   // tensor_dim0 hi=0 | tensor_dim1[15:0]=128
  g1[3] = (int)(64u << 16);           // tensor_dim1 hi=0 | tile_dim0=64
  g1[4] = (int)(uint32_t)C;           // tile_dim1=128, tile_dim2=0 (2D)
  g1[5] = (int)(uint32_t)L;           // tensor_dim0_stride = 8192 elements
  g1[6] = 0;
  g1[7] = 0;
  v4i z4 = {0, 0, 0, 0};
  v8i z8 = {0, 0, 0, 0, 0, 0, 0, 0};
  // clang-23 / therock-10.0 form: (g0, g1, g2, g3, g4, cpol) — trailing groups
  // unused for a 2D tile (tile_dim2/3/4 = 0), pass zeros.
  __builtin_amdgcn_tensor_load_to_lds(g0, g1, z4, z4, z8, 0);
}
static __device__ __forceinline__ uint32_t lds_byte_offset(const void* p) {
  // generic pointer to __shared__ = {aperture_hi, lds_byte_offset}
  return (uint32_t)(uintptr_t)p;
}
#endif

static __device__ __forceinline__ v8f wmma4(v2f a, v2f b, v8f c) {
  // D = A(16x4,f32) * B(4x16,f32) + C(16x16,f32)   -> v_wmma_f32_16x16x4_f32
  return __builtin_amdgcn_wmma_f32_16x16x4_f32(false, a, false, b, (short)0, c, false, false);
}

static __device__ __forceinline__ int enc_f32(float x) {
  int m = __float_as_int(x);
  return m >= 0 ? m : (m ^ 0x7FFFFFFF);   // monotone float -> signed-int key
}
static __device__ __forceinline__ float dec_f32(int k) {
  int m = k >= 0 ? k : (k ^ 0x7FFFFFFF);
  return __int_as_float(m);
}

// ============================================================
// Kernel 1: x11[b,c] = mean(|x[b,c,:]|)    (one block per (b,c) row)
// ============================================================
__global__ __launch_bounds__(256) void k1_absmean(const float* __restrict__ x,
                                                  float* __restrict__ x11) {
  const int row = blockIdx.x;                 // row = b*C + c
  const int tid = threadIdx.x;
  const float* p = x + (size_t)row * L;
  float s = 0.f;
#pragma unroll
  for (int i = 0; i < 8; ++i) {
    float4 v = *(const float4*)(p + tid * 4 + 1024 * i);
    s += fabsf(v.x) + fabsf(v.y) + fabsf(v.z) + fabsf(v.w);
  }
  __shared__ float red[256];
  red[tid] = s;
  __syncthreads();
  for (int off = 128; off > 0; off >>= 1) {
    if (tid < off) red[tid] += red[tid + off];
    __syncthreads();
  }
  if (tid == 0) x11[row] = red[0] * (1.0f / (float)L);
}

// ============================================================
// Kernel 2: channel-attention MLP -> T = x11 * sigmoid(...)
//           + init sum/max accumulators for the x1 stats pass
// ============================================================
__global__ __launch_bounds__(128) void k2_mlp(const float* __restrict__ x11,
                                              const float* __restrict__ wfc1,
                                              const float* __restrict__ bfc1,
                                              const float* __restrict__ g1,
                                              const float* __restrict__ b1,
                                              const float* __restrict__ rm1,
                                              const float* __restrict__ rv1,
                                              const float* __restrict__ wfc2,
                                              const float* __restrict__ bfc2,
                                              float* __restrict__ T,
                                              float* __restrict__ sum,
                                              int* __restrict__ maxenc) {
  const int b = blockIdx.x, tid = threadIdx.x;
  __shared__ float sx[C];
  __shared__ float sh[CQ];
  sx[tid] = x11[b * C + tid];
  // per-call re-init of stats accumulators (graph-replay safe)
  sum[b * C + tid] = 0.f;
  maxenc[b * C + tid] = INT_MIN;
  __syncthreads();
  if (tid < CQ) {
    float h = bfc1[tid];
#pragma unroll 8
    for (int c = 0; c < C; ++c) h += sx[c] * wfc1[tid * C + c];
    h = (h - rm1[tid]) * rsqrtf(rv1[tid] + EPS) * g1[tid] + b1[tid];
    sh[tid] = fmaxf(h, 0.f);
  }
  __syncthreads();
  float z = bfc2[tid];
#pragma unroll
  for (int j = 0; j < CQ; ++j) z += sh[j] * wfc2[tid * CQ + j];
  float x12 = 1.0f / (1.0f + expf(-z));
  T[b * C + tid] = sx[tid] * x12;
}

// ============================================================
// Kernel 3: recompute x0w = W1*X per tile (WMMA), form x1,
//           accumulate per-(b,o) sum & max over L via atomics.
// ============================================================
__global__ __launch_bounds__(256) void k3_x1stats(const float* __restrict__ x,
                                                  const float* __restrict__ w1,
                                                  const float* __restrict__ T,
                                                  float* __restrict__ sum,
                                                  int* __restrict__ maxenc) {
  const int b  = blockIdx.y;
  const int l0 = blockIdx.x * TILE;
  const int tid  = threadIdx.x;
  const int wave = tid >> 5, lane = tid & 31;
  const int half = lane >> 4, l16 = lane & 15;
  const int m0 = 16 * wave;

  __shared__ float xs[C * STRIDE];          // 36,864 B

  const float* xb = x + (size_t)b * C * L;
#ifdef USE_TDM
  if (tid < 32) {                            // wave 0 issues the DMA for the block
    tdm_load_tile(xb + l0, lds_byte_offset(&xs[0]));
    __builtin_amdgcn_s_wait_tensorcnt(0);
  }
#else
#pragma unroll
  for (int i = 0; i < 8; ++i) {
    int g = tid + 256 * i;
    int c = g >> 4, n4 = g & 15;
    float4 v = *(const float4*)(xb + (size_t)c * L + l0 + n4 * 4);
    *(float4*)&xs[c * STRIDE + n4 * 4] = v;
  }
#endif
  __syncthreads();

  v8f acc[SUB];
#pragma unroll
  for (int j = 0; j < SUB; ++j) acc[j] = (v8f){0.f,0.f,0.f,0.f,0.f,0.f,0.f,0.f};

  const int m = m0 + l16;
  for (int k0 = 0; k0 < C; k0 += 4) {
    const int krow = k0 + 2 * half;
    v2f af; af.x = w1[m * C + krow]; af.y = w1[m * C + krow + 1];
#pragma unroll
    for (int j = 0; j < SUB; ++j) {
      const int n = 16 * j + l16;
      v2f bf; bf.x = xs[krow * STRIDE + n]; bf.y = xs[(krow + 1) * STRIDE + n];
      acc[j] = wmma4(af, bf, acc[j]);
    }
  }

  const float* Tb = T + b * C;
#pragma unroll
  for (int r = 0; r < 8; ++r) {
    const int row = m0 + r + 8 * half;
    const float Tv = Tb[row];
    float s = 0.f, mx = -INFINITY;
#pragma unroll
    for (int j = 0; j < SUB; ++j) {
      float v   = acc[j][r];
      float xc  = xs[row * STRIDE + 16 * j + l16];
      float x1v = copysignf(fmaxf(fabsf(v) - Tv, 0.f), v) + xc;
      s += x1v;
      mx = fmaxf(mx, x1v);
    }
    for (int off = 8; off >= 1; off >>= 1) {
      s  += __shfl_xor(s, off, 32);
      mx  = fmaxf(mx, __shfl_xor(mx, off, 32));
    }
    if (l16 == 0) {
      atomicAdd(&sum[b * C + row], s);
      atomicMax(&maxenc[b * C + row], enc_f32(mx));
    }
  }
}

// ============================================================
// Kernel 4: spatial attention a[b,c] = sigmoid(conv_k3([mean;max]))
// ============================================================
__global__ __launch_bounds__(128) void k4_spatial(const float* __restrict__ sum,
                                                  const int* __restrict__ maxenc,
                                                  const float* __restrict__ wsp,
                                                  float* __restrict__ att) {
  const int b = blockIdx.x, c = threadIdx.x;
  __shared__ float pm[C + 2], px[C + 2];
  pm[c + 1] = sum[b * C + c] * (1.0f / (float)L);
  px[c + 1] = dec_f32(maxenc[b * C + c]);
  if (c < 2) { pm[c * (C + 1)] = 0.f; px[c * (C + 1)] = 0.f; }
  __syncthreads();
  float z = 0.f;
#pragma unroll
  for (int k = 0; k < 3; ++k)
    z += wsp[k] * pm[c + k] + wsp[3 + k] * px[c + k];
  att[b * C + c] = 1.0f / (1.0f + expf(-z));
}

// ============================================================
// Kernel 5: fused main pass.
//   sc  = Wsc (*) x   (3 shifted WMMA GEMMs, halo tile in LDS)
//   x0w = W1 * x      -> x1 written in-place into the LDS tile
//   x2  = W2 * x1
//   out = relu(BN2(sc) + x2 * a)
// ============================================================
__global__ __launch_bounds__(256) void k5_fused(const float* __restrict__ x,
                                                const float* __restrict__ w1,
                                                const float* __restrict__ w2,
                                                const float* __restrict__ wsc,
                                                const float* __restrict__ T,
                                                const float* __restrict__ att,
                                                const float* __restrict__ g2,
                                                const float* __restrict__ b2,
                                                const float* __restrict__ rm2,
                                                const float* __restrict__ rv2,
                                                float* __restrict__ out) {
  const int b  = blockIdx.y;
  const int l0 = blockIdx.x * TILE;
  const int tid  = threadIdx.x;
  const int wave = tid >> 5, lane = tid & 31;
  const int half = lane >> 4, l16 = lane & 15;
  const int m0 = 16 * wave;

  __shared__ float xs[C * STRIDE];   // cols: 0 = l0-1 (halo), 1..64 interior, 65 = l0+64 (halo)

  const float* xb = x + (size_t)b * C * L;
#ifdef USE_TDM
  if (tid < 32) {                    // wave 0: TDM DMA of the interior tile into cols 1..64
    tdm_load_tile(xb + l0, lds_byte_offset(&xs[1]));
    __builtin_amdgcn_s_wait_tensorcnt(0);
  }
#else
#pragma unroll
  for (int i = 0; i < 8; ++i) {
    int g = tid + 256 * i;
    int c = g >> 4, n4 = g & 15;
    float4 v = *(const float4*)(xb + (size_t)c * L + l0 + n4 * 4);
    float* dst = &xs[c * STRIDE + 1 + n4 * 4];
    dst[0] = v.x; dst[1] = v.y; dst[2] = v.z; dst[3] = v.w;
  }
#endif
  { // halo columns (bounds-checked, zero pad at sequence edges)
    int c = tid >> 1, side = tid & 1;
    int hcol = side ? (TILE + 1) : 0;
    int l = l0 - 1 + (side ? (TILE + 1) : 0);
    float v = (l >= 0 && l < L) ? xb[(size_t)c * L + l] : 0.f;
    xs[c * STRIDE + hcol] = v;
  }
  __syncthreads();

  const int m = m0 + l16;
  const v8f vz = (v8f){0.f,0.f,0.f,0.f,0.f,0.f,0.f,0.f};

  // ---- shortcut conv (3 shifted GEMMs) ----
  v8f scacc[SUB];
#pragma unroll
  for (int j = 0; j < SUB; ++j) scacc[j] = vz;
#pragma unroll
  for (int kk = 0; kk < 3; ++kk) {
    for (int k0 = 0; k0 < C; k0 += 4) {
      const int krow = k0 + 2 * half;
      v2f af;
      af.x = wsc[(m * C + krow) * 3 + kk];
      af.y = wsc[(m * C + krow + 1) * 3 + kk];
#pragma unroll
      for (int j = 0; j < SUB; ++j) {
        const int col = 16 * j + l16 + kk;   // interior col (n+1) shifted by kk-1
        v2f bf; bf.x = xs[krow * STRIDE + col]; bf.y = xs[(krow + 1) * STRIDE + col];
        scacc[j] = wmma4(af, bf, scacc[j]);
      }
    }
  }

  // ---- x0w = W1 * x ----
  v8f xw[SUB];
#pragma unroll
  for (int j = 0; j < SUB; ++j) xw[j] = vz;
  for (int k0 = 0; k0 < C; k0 += 4) {
    const int krow = k0 + 2 * half;
    v2f af; af.x = w1[m * C + krow]; af.y = w1[m * C + krow + 1];
#pragma unroll
    for (int j = 0; j < SUB; ++j) {
      const int col = 16 * j + l16 + 1;
      v2f bf; bf.x = xs[krow * STRIDE + col]; bf.y = xs[(krow + 1) * STRIDE + col];
      xw[j] = wmma4(af, bf, xw[j]);
    }
  }
  __syncthreads();     // all waves done reading x from LDS

  // ---- soft-threshold + residual: overwrite LDS interior with x1 ----
  const float* Tb = T + b * C;
#pragma unroll
  for (int r = 0; r < 8; ++r) {
    const int row = m0 + r + 8 * half;
    const float Tv = Tb[row];
#pragma unroll
    for (int j = 0; j < SUB; ++j) {
      const int col = 16 * j + l16 + 1;
      float v  = xw[j][r];
      float xc = xs[row * STRIDE + col];
      xs[row * STRIDE + col] = copysignf(fmaxf(fabsf(v) - Tv, 0.f), v) + xc;
    }
  }
  __syncthreads();     // x1 tile ready

  // ---- x2 = W2 * x1 ----
  v8f x2a[SUB];
#pragma unroll
  for (int j = 0; j < SUB; ++j) x2a[j] = vz;
  for (int k0 = 0; k0 < C; k0 += 4) {
    const int krow = k0 + 2 * half;
    v2f af; af.x = w2[m * C + krow]; af.y = w2[m * C + krow + 1];
#pragma unroll
    for (int j = 0; j < SUB; ++j) {
      const int col = 16 * j + l16 + 1;
      v2f bf; bf.x = xs[krow * STRIDE + col]; bf.y = xs[(krow + 1) * STRIDE + col];
      x2a[j] = wmma4(af, bf, x2a[j]);
    }
  }

  // ---- epilogue: BN(sc) + x2*a, ReLU, store ----
  float* ob = out + (size_t)b * C * L;
#pragma unroll
  for (int r = 0; r < 8; ++r) {
    const int row = m0 + r + 8 * half;
    const float sa  = att[b * C + row];
    const float inv = rsqrtf(rv2[row] + EPS);
    const float ga  = g2[row], bb = b2[row], rm = rm2[row];
#pragma unroll
    for (int j = 0; j < SUB; ++j) {
      const int n = 16 * j + l16;
      float o = (scacc[j][r] - rm) * inv * ga + bb + x2a[j][r] * sa;
      ob[(size_t)row * L + l0 + n] = fmaxf(o, 0.f);
    }
  }
}

// ============================================================
extern "C" void kernel_launch(void* const* d_in, const int* in_sizes, int n_in,
                              void* d_out, int out_size, void* d_ws, size_t ws_size,
                              hipStream_t stream) {
  const float* x    = (const float*)d_in[0];
  const float* wfc1 = (const float*)d_in[1];
  const float* bfc1 = (const float*)d_in[2];
  const float* g1   = (const float*)d_in[3];
  const float* b1   = (const float*)d_in[4];
  const float* rm1  = (const float*)d_in[5];
  const float* rv1  = (const float*)d_in[6];
  const float* wfc2 = (const float*)d_in[7];
  const float* bfc2 = (const float*)d_in[8];
  const float* w1   = (const float*)d_in[9];
  const float* w2   = (const float*)d_in[10];
  const float* wsp  = (const float*)d_in[11];
  const float* wsc  = (const float*)d_in[12];
  const float* g2   = (const float*)d_in[13];
  const float* b2   = (const float*)d_in[14];
  const float* rm2  = (const float*)d_in[15];
  const float* rv2  = (const float*)d_in[16];

  float* ws   = (float*)d_ws;
  float* x11  = ws;                     // [B*C]
  float* T    = ws + 1 * B * C;         // [B*C]
  float* sum  = ws + 2 * B * C;         // [B*C]
  int*   menc = (int*)(ws + 3 * B * C); // [B*C]
  float* att  = ws + 4 * B * C;         // [B*C]
  float* out  = (float*)d_out;

  k1_absmean<<<B * C, 256, 0, stream>>>(x, x11);
  k2_mlp<<<B, C, 0, stream>>>(x11, wfc1, bfc1, g1, b1, rm1, rv1, wfc2, bfc2, T, sum, menc);
  k3_x1stats<<<dim3(L / TILE, B), 256, 0, stream>>>(x, w1, T, sum, menc);
  k4_spatial<<<B, C, 0, stream>>>(sum, menc, wsp, att);
  k5_fused<<<dim3(L / TILE, B), 256, 0, stream>>>(x, w1, w2, wsc, T, att, g2, b2, rm2, rv2, out);
}